// NegLogLikelihood_45028437131761
// MI455X (gfx1250) — compile-verified
//
#include <hip/hip_runtime.h>

// ---------------------------------------------------------------------------
// NegLogLikelihood for MI455X (gfx1250, wave32).
//   mean, cov = split(output, 2, axis=0)   each [4096, 2048] f32
//   logdet    = sum(log(cov))              scalar
//   out       = -((mean-target)^2 / cov + logdet)
// Bandwidth-bound: ~128 MiB HBM traffic -> ~5.5 us at 23.3 TB/s.
// cov (32 MiB) is kept L2-resident (192 MB L2) between pass 1 and pass 3;
// mean/target/out are streamed with non-temporal hints.
// Wave reductions use V_WMMA_F32_16X16X4_F32 (B = ones) -> CDNA5 matrix path.
// ---------------------------------------------------------------------------

typedef float v2f __attribute__((ext_vector_type(2)));
typedef float v4f __attribute__((ext_vector_type(4)));
typedef float v8f __attribute__((ext_vector_type(8)));

#define BLOCK 256
#define WAVES_PER_BLOCK (BLOCK / 32)
#define RED_GRID 1024

// Full wave32 sum via WMMA with a ones B-matrix.
// A (16x4 f32, 2 VGPRs/lane): a[0] = x, a[1] = 0.
//   lane l<16  holds A[l][0], A[l][1];  lane l>=16 holds A[l-16][2], A[l-16][3]
// => D[m][n] = x_m + x_{m+16} for every n.
// Lane l<16 VGPR j = D[j][l];  lane l>=16 VGPR j = D[j+8][l-16].
// Sum d[0..7] => rows 0-7 total (lanes 0-15) / rows 8-15 total (lanes 16-31);
// one shfl_xor(16) completes the 32-lane sum (result in every lane).
// NOTE: EXEC must be all 1s at the call site.
__device__ __forceinline__ float wave_sum32(float x) {
    v2f a; a[0] = x;    a[1] = 0.0f;
    v2f b; b[0] = 1.0f; b[1] = 1.0f;
    v8f c = {};
    v8f d = __builtin_amdgcn_wmma_f32_16x16x4_f32(
        /*neg_a=*/false, a, /*neg_b=*/false, b,
        /*c_mod=*/(short)0, c, /*reuse_a=*/false, /*reuse_b=*/false);
    float s = ((d[0] + d[1]) + (d[2] + d[3])) + ((d[4] + d[5]) + (d[6] + d[7]));
    s += __shfl_xor(s, 16, 32);
    return s;
}

// Block-level sum (uniform control flow; valid only when blockDim.x == BLOCK
// and every thread participates). Result valid in thread 0.
__device__ __forceinline__ float block_sum(float acc) {
    __shared__ float wsum[WAVES_PER_BLOCK];
    float s = wave_sum32(acc);
    const int wid  = threadIdx.x >> 5;
    const int lane = threadIdx.x & 31;
    if (lane == 0) wsum[wid] = s;
    __syncthreads();
    float t = 0.0f;
    if (threadIdx.x == 0) {
#pragma unroll
        for (int i = 0; i < WAVES_PER_BLOCK; ++i) t += wsum[i];
    }
    return t;
}

// Pass 1: per-block partial sums of log(cov). cov read with default (RT)
// temporal hint so it stays resident in the 192 MB L2 for pass 3.
__global__ void logdet_partial_kernel(const v4f* __restrict__ cov4,
                                      float* __restrict__ partials, int n4) {
    float acc = 0.0f;
    const int stride = gridDim.x * blockDim.x;
    for (int i = blockIdx.x * blockDim.x + threadIdx.x; i < n4; i += stride) {
        v4f cv = cov4[i];                         // global_load_b128
        acc += (__logf(cv.x) + __logf(cv.y)) + (__logf(cv.z) + __logf(cv.w));
    }
    float t = block_sum(acc);
    if (threadIdx.x == 0) partials[blockIdx.x] = t;
}

// Pass 2: reduce RED_GRID partials to the final scalar.
__global__ void logdet_final_kernel(const float* __restrict__ partials,
                                    int n, float* __restrict__ result) {
    float acc = 0.0f;
    for (int i = threadIdx.x; i < n; i += blockDim.x) acc += partials[i];
    float t = block_sum(acc);
    if (threadIdx.x == 0) result[0] = t;
}

// Pass 3: out = -((mean-target)^2 / cov + logdet), 4 elems/thread.
// mean/target/out streamed non-temporal; cov regular (expected L2 hit).
__global__ void nll_elementwise_kernel(const v4f* __restrict__ mean4,
                                       const v4f* __restrict__ cov4,
                                       const v4f* __restrict__ tgt4,
                                       const float* __restrict__ logdet_p,
                                       v4f* __restrict__ out4, int n4) {
    const int i = blockIdx.x * blockDim.x + threadIdx.x;
    if (i >= n4) return;
    const float ld = logdet_p[0];
    v4f m = __builtin_nontemporal_load(&mean4[i]);
    v4f t = __builtin_nontemporal_load(&tgt4[i]);
    v4f c = cov4[i];
    v4f o;
    {
        float e;
        e = m.x - t.x;  o.x = -((e * e) / c.x + ld);
        e = m.y - t.y;  o.y = -((e * e) / c.y + ld);
        e = m.z - t.z;  o.z = -((e * e) / c.z + ld);
        e = m.w - t.w;  o.w = -((e * e) / c.w + ld);
    }
    __builtin_nontemporal_store(o, &out4[i]);
}

extern "C" void kernel_launch(void* const* d_in, const int* in_sizes, int n_in,
                              void* d_out, int out_size, void* d_ws, size_t ws_size,
                              hipStream_t stream) {
    const float* output = (const float*)d_in[0];   // [8192, 2048] = [mean; cov]
    const float* target = (const float*)d_in[1];   // [4096, 2048]
    const int N  = in_sizes[0] / 2;                // 4096*2048 = 8388608
    const int n4 = N / 4;                          // exact: N is 2^23

    const float* mean = output;
    const float* cov  = output + N;

    float* partials = (float*)d_ws;                // RED_GRID floats
    float* logdet   = partials + RED_GRID;         // 1 float

    logdet_partial_kernel<<<RED_GRID, BLOCK, 0, stream>>>(
        (const v4f*)cov, partials, n4);

    logdet_final_kernel<<<1, BLOCK, 0, stream>>>(partials, RED_GRID, logdet);

    const int eblocks = (n4 + BLOCK - 1) / BLOCK;
    nll_elementwise_kernel<<<eblocks, BLOCK, 0, stream>>>(
        (const v4f*)mean, (const v4f*)cov, (const v4f*)target,
        logdet, (v4f*)d_out, n4);
}